// Text2SemanticDecoder_66881230734020
// MI455X (gfx1250) — compile-verified
//
#include <hip/hip_runtime.h>
#include <hip/hip_bf16.h>

// Model dims (fixed by the reference)
#define L_LAYERS 24
#define DMODEL   512
#define NHEAD    16
#define HD       32
#define BQ       16
#define TMAXC    1024
#define DFF2     2048

typedef __attribute__((ext_vector_type(2))) float v2f;
typedef __attribute__((ext_vector_type(8))) float v8f;

// ---------------------------------------------------------------------------
// 16xK @ KxN fp32 GEMM using V_WMMA_F32_16X16X4_F32.
// One wave computes one 16x16 output tile; 4 waves / block.
// K, N are template constants -> all inner-loop addresses become
// base + immediate offset (24-bit IOFFSET), one pointer bump per 8 WMMAs.
// Per outer step: 24-load clause (8x b64 A + 16x b32 B), then 8 WMMAs.
//
// Operand layouts (ISA 7.12.2):
//  A (16x4):  lanes 0-15 hold (K=k0,k0+1), lanes 16-31 hold (K=k0+2,k0+3)
//  B (4x16):  VGPR0 = rows k0 / k0+2, VGPR1 = rows k0+1 / k0+3 (N = lane&15)
//  C/D(16x16): VGPR r = row r (lanes 0-15) and row r+8 (lanes 16-31)
// ---------------------------------------------------------------------------
template <int K, int N, bool RES, bool RELU>
__global__ __launch_bounds__(128) void gemm16_wmma(
    const float* __restrict__ X,    // [16, K] row-major
    const float* __restrict__ W,    // [K, N]  row-major
    const float* __restrict__ bias, // [N]
    const float* __restrict__ res,  // [16, N] (used iff RES)
    float* __restrict__ out)        // [16, N]
{
  const int lane = threadIdx.x & 31;
  const int wave = threadIdx.x >> 5;
  const int n0   = (blockIdx.x * 4 + wave) << 4;   // wave-uniform
  if (n0 >= N) return;                              // whole wave exits together

  const int hl = lane >> 4;   // 0: lanes 0-15, 1: lanes 16-31
  const int l  = lane & 15;

  const float* xp = X + l * K + 2 * hl;             // A stream (row l)
  const float* wp = W + (2 * hl) * N + n0 + l;      // B stream (col n0+l)

  v8f c = {};
  for (int kb = 0; kb < K; kb += 32) {              // 8 WMMAs per body
    v2f a[8], bv[8];
    #pragma unroll
    for (int u = 0; u < 8; ++u) {
      a[u] = *(const v2f*)(xp + 4 * u);             // global_load_b64, imm off
      bv[u].x = wp[4 * u * N];                      // global_load_b32, imm off
      bv[u].y = wp[4 * u * N + N];
    }
    if (kb + 32 < K)
      __builtin_prefetch(wp + 32 * N, 0, 1);        // global_prefetch_b8
    #pragma unroll
    for (int u = 0; u < 8; ++u)
      c = __builtin_amdgcn_wmma_f32_16x16x4_f32(
              /*neg_a=*/false, a[u], /*neg_b=*/false, bv[u],
              /*c_mod=*/(short)0, c, /*reuse_a=*/false, /*reuse_b=*/false);
    xp += 32;
    wp += 32 * N;
  }

  const int col = n0 + l;
  const float bcol = bias[col];
  #pragma unroll
  for (int r = 0; r < 8; ++r) {
    const int row = r + 8 * hl;
    const int idx = row * N + col;
    float v = c[r] + bcol;
    if (RES)  v += res[idx];
    if (RELU) v = fmaxf(v, 0.0f);
    out[idx] = v;
  }
}

// ---------------------------------------------------------------------------
// Attention for one (b,h): scores -> softmax -> weighted V sum.
// 128 threads (4 waves). New token K/V read from the QKV scratch (position
// t == kv_len) so the input caches are never mutated.
// ---------------------------------------------------------------------------
__global__ __launch_bounds__(128) void attn_kernel(
    const float* __restrict__ qkv,   // [16, 3*D]; cols: q | k_new | v_new
    const float* __restrict__ kc,    // [B,H,TMAX,HD] (this layer)
    const float* __restrict__ vc,    // [B,H,TMAX,HD] (this layer)
    const int*   __restrict__ kvlen, // [B]
    float* __restrict__ out)         // [16, D]
{
  __shared__ float s_sc[TMAXC];
  __shared__ float s_q[HD];
  __shared__ float s_red[16];
  __shared__ float s_o[4 * HD];

  const int bh = blockIdx.x;
  const int b = bh >> 4, h = bh & 15;
  const int tid = threadIdx.x, lane = tid & 31, wave = tid >> 5;
  const int tlen = kvlen[b] + 1;   // positions 0..kv_len inclusive

  if (tid < HD) s_q[tid] = qkv[(size_t)b * 3 * DMODEL + h * HD + tid];
  __syncthreads();

  const float* Kb   = kc + (size_t)bh * TMAXC * HD;
  const float* Vb   = vc + (size_t)bh * TMAXC * HD;
  const float* knew = qkv + (size_t)b * 3 * DMODEL +     DMODEL + h * HD;
  const float* vnew = qkv + (size_t)b * 3 * DMODEL + 2 * DMODEL + h * HD;
  const float scale = 0.1767766952966369f;  // 1/sqrt(32)

  // Pass 1: scores + running max (lane owns row t -> b128 streaming loads)
  float lmax = -1e30f;
  for (int t = tid; t < tlen; t += 128) {
    const float* kr = (t == tlen - 1) ? knew : (Kb + (size_t)t * HD);
    float d = 0.0f;
    #pragma unroll
    for (int i = 0; i < HD; ++i) d += s_q[i] * kr[i];
    d *= scale;
    s_sc[t] = d;
    lmax = fmaxf(lmax, d);
  }
  for (int off = 16; off; off >>= 1) lmax = fmaxf(lmax, __shfl_xor(lmax, off, 32));
  if (lane == 0) s_red[wave] = lmax;
  __syncthreads();
  if (tid == 0) {
    float m = s_red[0];
    for (int w = 1; w < 4; ++w) m = fmaxf(m, s_red[w]);
    s_red[0] = m;
  }
  __syncthreads();
  const float gmax = s_red[0];

  // Pass 2: exp + sum
  float lsum = 0.0f;
  for (int t = tid; t < tlen; t += 128) {
    float w = __expf(s_sc[t] - gmax);
    s_sc[t] = w;
    lsum += w;
  }
  for (int off = 16; off; off >>= 1) lsum += __shfl_xor(lsum, off, 32);
  if (lane == 0) s_red[8 + wave] = lsum;
  __syncthreads();
  if (tid == 0) {
    float s2 = 0.0f;
    for (int w = 0; w < 4; ++w) s2 += s_red[8 + w];
    s_red[8] = s2;
  }
  __syncthreads();
  const float inv = 1.0f / s_red[8];

  // Pass 3: lane owns output dim d=lane -> V rows read fully coalesced
  float o = 0.0f;
  for (int t = wave; t < tlen; t += 4) {
    const float* vr = (t == tlen - 1) ? vnew : (Vb + (size_t)t * HD);
    o += s_sc[t] * vr[lane];
  }
  s_o[wave * HD + lane] = o;
  __syncthreads();
  if (wave == 0) {
    float r = s_o[lane] + s_o[HD + lane] + s_o[2 * HD + lane] + s_o[3 * HD + lane];
    out[(size_t)b * DMODEL + h * HD + lane] = r * inv;
  }
}

// ---------------------------------------------------------------------------
// LayerNorm over D=512, one block per row, 256 threads x 2 elems.
// ---------------------------------------------------------------------------
__global__ __launch_bounds__(256) void ln_kernel(
    const float* __restrict__ in, const float* __restrict__ g,
    const float* __restrict__ b, float* __restrict__ out)
{
  __shared__ float red[8];
  const int row = blockIdx.x, tid = threadIdx.x;
  const int lane = tid & 31, wave = tid >> 5;

  const float x0 = in[(size_t)row * DMODEL + tid];
  const float x1 = in[(size_t)row * DMODEL + 256 + tid];

  float s = x0 + x1;
  for (int off = 16; off; off >>= 1) s += __shfl_xor(s, off, 32);
  if (lane == 0) red[wave] = s;
  __syncthreads();
  if (tid == 0) { float t = 0; for (int w = 0; w < 8; ++w) t += red[w]; red[0] = t; }
  __syncthreads();
  const float mu = red[0] * (1.0f / DMODEL);
  __syncthreads();

  const float d0 = x0 - mu, d1 = x1 - mu;
  float v = d0 * d0 + d1 * d1;
  for (int off = 16; off; off >>= 1) v += __shfl_xor(v, off, 32);
  if (lane == 0) red[wave] = v;
  __syncthreads();
  if (tid == 0) { float t = 0; for (int w = 0; w < 8; ++w) t += red[w]; red[0] = t; }
  __syncthreads();
  const float rstd = rsqrtf(red[0] * (1.0f / DMODEL) + 1e-5f);

  out[(size_t)row * DMODEL + tid]       = d0 * rstd * g[tid]       + b[tid];
  out[(size_t)row * DMODEL + 256 + tid] = d1 * rstd * g[256 + tid] + b[256 + tid];
}

// ---------------------------------------------------------------------------
extern "C" void kernel_launch(void* const* d_in, const int* in_sizes, int n_in,
                              void* d_out, int out_size, void* d_ws, size_t ws_size,
                              hipStream_t stream) {
  const float* x       = (const float*)d_in[0];
  const float* k_cache = (const float*)d_in[1];
  const float* v_cache = (const float*)d_in[2];
  const float* ln1_g   = (const float*)d_in[3];
  const float* ln1_b   = (const float*)d_in[4];
  const float* qkv_w   = (const float*)d_in[5];
  const float* qkv_b   = (const float*)d_in[6];
  const float* out_w   = (const float*)d_in[7];
  const float* out_b   = (const float*)d_in[8];
  const float* ln2_g   = (const float*)d_in[9];
  const float* ln2_b   = (const float*)d_in[10];
  const float* mlp_w1  = (const float*)d_in[11];
  const float* mlp_b1  = (const float*)d_in[12];
  const float* mlp_w2  = (const float*)d_in[13];
  const float* mlp_b2  = (const float*)d_in[14];
  // d_in[15] attn_mask, d_in[17] batch_indices: implied by kv_cache_len
  const int*   kvlen   = (const int*)d_in[16];

  float* ws  = (float*)d_ws;
  float* H0  = ws;              // [16,512]   current hidden state
  float* H1  = ws + 8192;       // [16,512]   post-LN1 hidden
  float* QKV = ws + 16384;      // [16,1536]
  float* O   = ws + 40960;      // [16,512]   attention output
  float* T   = ws + 49152;      // [16,512]   pre-LN temp
  float* FF  = ws + 57344;      // [16,2048]  MLP hidden

  hipMemcpyAsync(H0, x, (size_t)BQ * DMODEL * sizeof(float),
                 hipMemcpyDeviceToDevice, stream);

  const size_t kv_layer = (size_t)BQ * NHEAD * TMAXC * HD;

  for (int l = 0; l < L_LAYERS; ++l) {
    const float* Wqkv = qkv_w  + (size_t)l * DMODEL * 3 * DMODEL;
    const float* bqkv = qkv_b  + (size_t)l * 3 * DMODEL;
    const float* Wo   = out_w  + (size_t)l * DMODEL * DMODEL;
    const float* bo   = out_b  + (size_t)l * DMODEL;
    const float* W1   = mlp_w1 + (size_t)l * DMODEL * DFF2;
    const float* bm1  = mlp_b1 + (size_t)l * DFF2;
    const float* W2   = mlp_w2 + (size_t)l * DFF2 * DMODEL;
    const float* bm2  = mlp_b2 + (size_t)l * DMODEL;
    const float* g1   = ln1_g + (size_t)l * DMODEL;
    const float* be1  = ln1_b + (size_t)l * DMODEL;
    const float* g2   = ln2_g + (size_t)l * DMODEL;
    const float* be2  = ln2_b + (size_t)l * DMODEL;
    const float* kc   = k_cache + (size_t)l * kv_layer;
    const float* vc   = v_cache + (size_t)l * kv_layer;

    // qkv = h @ Wqkv + b            [16,1536]
    gemm16_wmma<DMODEL, 3 * DMODEL, false, false>
        <<<(3 * DMODEL) / 64, 128, 0, stream>>>(H0, Wqkv, bqkv, nullptr, QKV);
    // attention (KV cache + fresh token from QKV)
    attn_kernel<<<BQ * NHEAD, 128, 0, stream>>>(QKV, kc, vc, kvlen, O);
    // T = h + o @ Wo + bo
    gemm16_wmma<DMODEL, DMODEL, true, false>
        <<<DMODEL / 64, 128, 0, stream>>>(O, Wo, bo, H0, T);
    // H1 = LN1(T)
    ln_kernel<<<BQ, 256, 0, stream>>>(T, g1, be1, H1);
    // FF = relu(H1 @ W1 + b1)
    gemm16_wmma<DMODEL, DFF2, false, true>
        <<<DFF2 / 64, 128, 0, stream>>>(H1, W1, bm1, nullptr, FF);
    // T = H1 + FF @ W2 + b2
    gemm16_wmma<DFF2, DMODEL, true, false>
        <<<DMODEL / 64, 128, 0, stream>>>(FF, W2, bm2, H1, T);
    // next h = LN2(T); last layer writes d_out directly
    float* dst = (l == L_LAYERS - 1) ? (float*)d_out : H0;
    ln_kernel<<<BQ, 256, 0, stream>>>(T, g2, be2, dst);
  }
}